// cal_ssim_loss_73375221285077
// MI455X (gfx1250) — compile-verified
//
#include <hip/hip_runtime.h>

typedef float v2f __attribute__((ext_vector_type(2)));
typedef float v8f __attribute__((ext_vector_type(8)));

#define IMG   512
#define TILE  32
#define IN_R  48      // input tile rows in LDS (with halo + roundup)
#define IN_C  44      // input tile cols in LDS (with halo + roundup)
#define HST   18      // hstage row stride in floats (bank-conflict free)
#define HROWS 32      // hconv rows staged per quadrant (26 used, 2 row-blocks)
#define NPIX  12582912.0f   // 16*3*512*512

// 11-tap Gaussian, sigma=1.5, normalized (matches reference make_window()).
__constant__ float GK[11] = {
    0.00102838f, 0.00759876f, 0.03600077f, 0.10936113f, 0.21300553f,
    0.26601172f, 0.21300553f, 0.10936113f, 0.03600077f, 0.00759876f,
    0.00102838f};

__device__ __forceinline__ float gband(int d) {
    int dc = d < 0 ? 0 : (d > 10 ? 10 : d);
    return (d >= 0 && d <= 10) ? GK[dc] : 0.0f;
}

__global__ __launch_bounds__(128) void ssim_main_kernel(
    const float* __restrict__ X, const float* __restrict__ Y,
    float* __restrict__ partials)
{
    __shared__ float sX[IN_R * IN_C];
    __shared__ float sY[IN_R * IN_C];
    __shared__ float sH[4 * 5 * HROWS * HST];
    __shared__ float sWsum[4];

    const int tid  = threadIdx.x;
    const int lane = tid & 31;
    const int wave = tid >> 5;
    const int r0   = blockIdx.y * TILE;
    const int c0   = blockIdx.x * TILE;
    const size_t base = (size_t)blockIdx.z * (size_t)(IMG * IMG);

    // Cooperative haloed tile load with zero padding at image borders.
    for (int i = tid; i < IN_R * IN_C; i += 128) {
        int rr = i / IN_C;
        int cc = i - rr * IN_C;
        int gr = r0 - 5 + rr;
        int gc = c0 - 5 + cc;
        bool ok = ((unsigned)gr < (unsigned)IMG) && ((unsigned)gc < (unsigned)IMG);
        size_t g = base + (size_t)(ok ? gr : 0) * IMG + (size_t)(ok ? gc : 0);
        sX[i] = ok ? X[g] : 0.0f;
        sY[i] = ok ? Y[g] : 0.0f;
    }
    __syncthreads();

    // Per-lane banded Gaussian weight chunks.  Identical register image for
    // the horizontal pass (B operand, W[i][n]=g[i-n]) and the vertical pass
    // (A operand, Wv[m][t]=g[t-m]) given the symmetric f32 A/B VGPR layouts.
    const int half = lane >> 4;   // 0: K 0,1   1: K 2,3
    const int mn   = lane & 15;   // M (A) or N (B)
    const int koff = half * 2;
    v2f wt[7];
#pragma unroll
    for (int k = 0; k < 7; ++k) {
        int d0 = 4 * k + koff - mn;
        wt[k].x = gband(d0);
        wt[k].y = gband(d0 + 1);
    }

    const int qr = (wave >> 1) * 16;   // quadrant row offset in tile
    const int qc = (wave & 1) * 16;    // quadrant col offset in tile

    // ---- Horizontal pass: 2 row-blocks x 7 K-chunks x 5 quantities ----
    v8f hacc[5][2];
#pragma unroll
    for (int q = 0; q < 5; ++q) {
#pragma unroll
        for (int rb = 0; rb < 2; ++rb) {
            v8f z = {0.f, 0.f, 0.f, 0.f, 0.f, 0.f, 0.f, 0.f};
            hacc[q][rb] = z;
        }
    }

#pragma unroll
    for (int rb = 0; rb < 2; ++rb) {
        const int row = qr + rb * 16 + mn;   // LDS input row (A-layout M)
#pragma unroll
        for (int k = 0; k < 7; ++k) {
            const int col = qc + 4 * k + koff;       // even -> 8B aligned
            v2f x = *(const v2f*)&sX[row * IN_C + col];
            v2f y = *(const v2f*)&sY[row * IN_C + col];
            v2f xx = x * x;
            v2f yy = y * y;
            v2f xy = x * y;
            hacc[0][rb] = __builtin_amdgcn_wmma_f32_16x16x4_f32(
                false, x,  false, wt[k], (short)0, hacc[0][rb], false, false);
            hacc[1][rb] = __builtin_amdgcn_wmma_f32_16x16x4_f32(
                false, y,  false, wt[k], (short)0, hacc[1][rb], false, false);
            hacc[2][rb] = __builtin_amdgcn_wmma_f32_16x16x4_f32(
                false, xx, false, wt[k], (short)0, hacc[2][rb], false, false);
            hacc[3][rb] = __builtin_amdgcn_wmma_f32_16x16x4_f32(
                false, yy, false, wt[k], (short)0, hacc[3][rb], false, false);
            hacc[4][rb] = __builtin_amdgcn_wmma_f32_16x16x4_f32(
                false, xy, false, wt[k], (short)0, hacc[4][rb], false, false);
        }
    }

    // ---- Stage hconv (D layout -> LDS rows) for the vertical B operand ----
    const int hbase = wave * (5 * HROWS * HST);
#pragma unroll
    for (int q = 0; q < 5; ++q) {
#pragma unroll
        for (int rb = 0; rb < 2; ++rb) {
#pragma unroll
            for (int e = 0; e < 8; ++e) {
                int t = rb * 16 + e + half * 8;   // D: lanes16-31 hold M=e+8
                sH[hbase + q * (HROWS * HST) + t * HST + mn] = hacc[q][rb][e];
            }
        }
    }
    // hstage is private per wave; compiler-inserted DScnt waits order the
    // stores against the loads below (no workgroup barrier needed).

    // ---- Vertical pass: 5 quantities x 7 K-chunks ----
    v8f vacc[5];
#pragma unroll
    for (int q = 0; q < 5; ++q) {
        v8f z = {0.f, 0.f, 0.f, 0.f, 0.f, 0.f, 0.f, 0.f};
        vacc[q] = z;
    }
#pragma unroll
    for (int q = 0; q < 5; ++q) {
        const int qb = hbase + q * (HROWS * HST);
#pragma unroll
        for (int k = 0; k < 7; ++k) {
            const int t0 = 4 * k + koff;
            v2f b;
            b.x = sH[qb + t0 * HST + mn];
            b.y = sH[qb + (t0 + 1) * HST + mn];
            vacc[q] = __builtin_amdgcn_wmma_f32_16x16x4_f32(
                false, wt[k], false, b, (short)0, vacc[q], false, false);
        }
    }

    // ---- SSIM map + deterministic reduction ----
    const float C1 = 1e-4f, C2 = 9e-4f;
    float ssum = 0.0f;
#pragma unroll
    for (int e = 0; e < 8; ++e) {
        float m1 = vacc[0][e], m2 = vacc[1][e];
        float m1s = m1 * m1, m2s = m2 * m2, m12 = m1 * m2;
        float s1  = vacc[2][e] - m1s;
        float s2  = vacc[3][e] - m2s;
        float s12 = vacc[4][e] - m12;
        float num = (2.0f * m12 + C1) * (2.0f * s12 + C2);
        float den = (m1s + m2s + C1) * (s1 + s2 + C2);
        ssum += num / den;
    }
#pragma unroll
    for (int off = 16; off > 0; off >>= 1)
        ssum += __shfl_down(ssum, off, 32);
    if (lane == 0) sWsum[wave] = ssum;
    __syncthreads();
    if (tid == 0) {
        int bid = ((int)blockIdx.z * (int)gridDim.y + (int)blockIdx.y) *
                      (int)gridDim.x + (int)blockIdx.x;
        partials[bid] = (sWsum[0] + sWsum[1]) + (sWsum[2] + sWsum[3]);
    }
}

__global__ __launch_bounds__(256) void ssim_reduce_kernel(
    const float* __restrict__ partials, int n, float* __restrict__ out)
{
    __shared__ float sm[256];
    const int tid = threadIdx.x;
    float s = 0.0f;
    for (int i = tid; i < n; i += 256) s += partials[i];   // fixed order
    sm[tid] = s;
    __syncthreads();
    for (int st = 128; st > 0; st >>= 1) {
        if (tid < st) sm[tid] += sm[tid + st];
        __syncthreads();
    }
    if (tid == 0) out[0] = 1.0f - sm[0] / NPIX;
}

extern "C" void kernel_launch(void* const* d_in, const int* in_sizes, int n_in,
                              void* d_out, int out_size, void* d_ws, size_t ws_size,
                              hipStream_t stream)
{
    const float* X = (const float*)d_in[0];   // infer (16,3,512,512) f32
    const float* Y = (const float*)d_in[1];   // gt    (16,3,512,512) f32
    float* out = (float*)d_out;
    float* partials = (float*)d_ws;           // 16*16*48 = 12288 floats

    dim3 grid(IMG / TILE, IMG / TILE, 16 * 3); // 16 x 16 x 48
    dim3 block(128);
    ssim_main_kernel<<<grid, block, 0, stream>>>(X, Y, partials);
    ssim_reduce_kernel<<<1, 256, 0, stream>>>(partials, 16 * 16 * 48, out);
}